// GlobalAttentionPooling_56839597195672
// MI455X (gfx1250) — compile-verified
//
#include <hip/hip_runtime.h>
#include <math.h>

// GlobalAttentionPooling for MI455X (gfx1250, wave32).
// Bandwidth-bound: ~1.02 GB of z traffic -> ~44us floor @ 23.3 TB/s.
// Segment-weighted-sum uses v_wmma_f32_16x16x4_f32 (full f32 precision).

typedef __attribute__((ext_vector_type(2))) float v2f;
typedef __attribute__((ext_vector_type(8))) float v8f;

#define GB        2048          // blocks for streaming passes
#define BT        256           // threads per block (8 waves on wave32)
#define DIMS      128
#define NGRAPH    4096
#define NEG_FLT_MAX (-3.402823466e+38f)

// ---------------- Pass 1: scores + per-block max ----------------
// wave-per-node: lane l loads z[n][4l..4l+3] (coalesced b128), butterfly reduce.
__global__ void score_max_kernel(const float* __restrict__ z,
                                 const float* __restrict__ W,
                                 const float* __restrict__ b,
                                 float* __restrict__ s,
                                 float* __restrict__ blockMax,
                                 int N) {
    __shared__ float red[BT];
    const int tid  = threadIdx.x;
    const int lane = tid & 31;
    const int wave = tid >> 5;
    const int wavesPerBlock = BT / 32;

    const int chunk = (N + (int)gridDim.x - 1) / (int)gridDim.x;
    const int begin = (int)blockIdx.x * chunk;
    const int finish = (begin + chunk < N) ? (begin + chunk) : N;

    const float4 w4  = reinterpret_cast<const float4*>(W)[lane];
    const float bval = b[0];

    float lmax = NEG_FLT_MAX;
    for (int n = begin + wave; n < finish; n += wavesPerBlock) {
        float4 zv = reinterpret_cast<const float4*>(z)[(size_t)n * 32 + lane];
        float dot = zv.x * w4.x + zv.y * w4.y + zv.z * w4.z + zv.w * w4.w;
        #pragma unroll
        for (int off = 16; off > 0; off >>= 1)
            dot += __shfl_xor(dot, off, 32);
        const float sc = dot + bval;
        if (lane == 0) s[n] = sc;
        lmax = fmaxf(lmax, sc);
    }
    red[tid] = lmax; __syncthreads();
    #pragma unroll
    for (int st = BT / 2; st > 0; st >>= 1) {
        if (tid < st) red[tid] = fmaxf(red[tid], red[tid + st]);
        __syncthreads();
    }
    if (tid == 0) blockMax[blockIdx.x] = red[0];
}

// ---------------- Pass 2: global max (single block) ----------------
__global__ void reduce_max_kernel(const float* __restrict__ part, int P,
                                  float* __restrict__ gsc) {
    __shared__ float red[BT];
    const int tid = threadIdx.x;
    float v = NEG_FLT_MAX;
    for (int i = tid; i < P; i += BT) v = fmaxf(v, part[i]);
    red[tid] = v; __syncthreads();
    #pragma unroll
    for (int st = BT / 2; st > 0; st >>= 1) {
        if (tid < st) red[tid] = fmaxf(red[tid], red[tid + st]);
        __syncthreads();
    }
    if (tid == 0) gsc[0] = red[0];
}

// ---------------- Pass 3: e = exp(s - max), per-block sums (fixed order) ---
__global__ void exp_sum_kernel(float* __restrict__ s,
                               const float* __restrict__ gsc,
                               float* __restrict__ blockSum, int N) {
    __shared__ float red[BT];
    const int tid = threadIdx.x;
    const int chunk = (N + (int)gridDim.x - 1) / (int)gridDim.x;
    const int begin = (int)blockIdx.x * chunk;
    const int finish = (begin + chunk < N) ? (begin + chunk) : N;
    const float gmax = gsc[0];

    float lsum = 0.0f;
    for (int i = begin + tid; i < finish; i += BT) {
        const float e = expf(s[i] - gmax);
        s[i] = e;               // overwrite scores with unnormalized weights
        lsum += e;
    }
    red[tid] = lsum; __syncthreads();
    #pragma unroll
    for (int st = BT / 2; st > 0; st >>= 1) {
        if (tid < st) red[tid] += red[tid + st];
        __syncthreads();
    }
    if (tid == 0) blockSum[blockIdx.x] = red[0];
}

// ---------------- Pass 4: global sum (single block, deterministic) --------
__global__ void reduce_sum_kernel(const float* __restrict__ part, int P,
                                  float* __restrict__ gsc) {
    __shared__ float red[BT];
    const int tid = threadIdx.x;
    float v = 0.0f;
    for (int i = tid; i < P; i += BT) v += part[i];
    red[tid] = v; __syncthreads();
    #pragma unroll
    for (int st = BT / 2; st > 0; st >>= 1) {
        if (tid < st) red[tid] += red[tid + st];
        __syncthreads();
    }
    if (tid == 0) gsc[1] = red[0];
}

// ---------------- Pass 5: per-graph weighted sum via WMMA -----------------
__device__ __forceinline__ int lower_bound_i(const int* __restrict__ a, int n, int v) {
    int lo = 0, hi = n;
    while (lo < hi) { int m = (lo + hi) >> 1; if (a[m] < v) lo = m + 1; else hi = m; }
    return lo;
}

// One wave (32 threads) per graph. batch_index is sorted -> binary search
// gives [start,end). Segment sum  sums[d] = sum_i w_i * z[i][d]  is mapped to
// v_wmma_f32_16x16x4_f32: K=4 nodes per step, 8 tiles cover the 128 dims.
//   A (16x4):  VGPR0 = K0|K2 (lane<16 | lane>=16), VGPR1 = K1|K3, rows replicated.
//   B (4x16):  VGPR0 = K0|K2 row striped over lanes, VGPR1 = K1|K3.
//   D row M=0 lives in element 0 of the 8-VGPR accumulator, lanes 0..15.
__global__ void pool_wmma_kernel(const float* __restrict__ z,
                                 const int*   __restrict__ bidx,
                                 const float* __restrict__ wts,
                                 const float* __restrict__ gsc,
                                 float* __restrict__ out,
                                 int N) {
    const int g      = blockIdx.x;
    const int lane   = threadIdx.x;   // 0..31, wave32
    const int half   = lane >> 4;     // 0: K0/K1 lanes, 1: K2/K3 lanes
    const int lanelo = lane & 15;

    const int start = lower_bound_i(bidx, N, g);
    const int end   = lower_bound_i(bidx, N, g + 1);
    const int count = end - start;

    v8f acc[8] = {};                  // 8 dim-tiles of 16 columns each

    for (int k0 = start; k0 < end; k0 += 4) {
        const int n0 = k0 + half * 2; // node for K = 2*half
        const int n1 = n0 + 1;        // node for K = 2*half + 1
        const int zn0 = (n0 < N - 1) ? n0 : (N - 1);
        const int zn1 = (n1 < N - 1) ? n1 : (N - 1);

        float a0 = wts[zn0]; if (n0 >= end) a0 = 0.0f;   // tail -> weight 0
        float a1 = wts[zn1]; if (n1 >= end) a1 = 0.0f;

        v2f A; A[0] = a0; A[1] = a1;

        const float* zr0 = z + (size_t)zn0 * DIMS + lanelo;
        const float* zr1 = z + (size_t)zn1 * DIMS + lanelo;
        #pragma unroll
        for (int t = 0; t < 8; ++t) {
            v2f B; B[0] = zr0[t * 16]; B[1] = zr1[t * 16];
            acc[t] = __builtin_amdgcn_wmma_f32_16x16x4_f32(
                false, A, false, B, (short)0, acc[t], false, false);
        }
    }

    const float cnt   = (count > 0) ? (float)count : 1.0f;
    const float scale = 1.0f / (gsc[1] * cnt);   // softmax denom * mean divisor
    if (lane < 16) {
        #pragma unroll
        for (int t = 0; t < 8; ++t)
            out[(size_t)g * DIMS + t * 16 + lanelo] = acc[t][0] * scale;
    }
}

// ---------------------------------------------------------------------------
extern "C" void kernel_launch(void* const* d_in, const int* in_sizes, int n_in,
                              void* d_out, int out_size, void* d_ws, size_t ws_size,
                              hipStream_t stream) {
    const float* z    = (const float*)d_in[0];  // [N,128] f32
    const int*   bidx = (const int*)  d_in[1];  // [N] sorted graph ids
    const float* W    = (const float*)d_in[2];  // [1,128] f32
    const float* b    = (const float*)d_in[3];  // [1] f32
    float* out = (float*)d_out;                 // [4096,128] f32

    const int N = in_sizes[1];                  // node count

    // workspace: N scores/weights + 2*GB partials + 2 scalars (~4 MB)
    float* sbuf  = (float*)d_ws;
    float* part1 = sbuf + N;
    float* part2 = part1 + GB;
    float* gsc   = part2 + GB;   // [0]=global max, [1]=global sum

    score_max_kernel <<<GB, BT, 0, stream>>>(z, W, b, sbuf, part1, N);
    reduce_max_kernel<<<1,  BT, 0, stream>>>(part1, GB, gsc);
    exp_sum_kernel   <<<GB, BT, 0, stream>>>(sbuf, gsc, part2, N);
    reduce_sum_kernel<<<1,  BT, 0, stream>>>(part2, GB, gsc);
    pool_wmma_kernel <<<NGRAPH, 32, 0, stream>>>(z, bidx, sbuf, gsc, out, N);
}